// QLSTM_65481071401579
// MI455X (gfx1250) — compile-verified
//
#include <hip/hip_runtime.h>

typedef float v2f __attribute__((ext_vector_type(2)));
typedef float v8f __attribute__((ext_vector_type(8)));

#define SEQ   256
#define BATCH 256
#define DDIM  512
#define HDIM  512
#define COMB  1024
#define HPAD  516   // padded LDS row stride for h tile (bank-conflict free b64 reads)
#define QPAD  36    // padded LDS row stride for angle/q tiles
#define TPB   256   // 8 wave32 per workgroup

__device__ __forceinline__ float sigm_(float x) { return 1.0f / (1.0f + __expf(-x)); }

// One persistent workgroup per 16-row batch tile; runs all SEQ timesteps.
__global__ __launch_bounds__(TPB) void qlstm_fused(
    const float* __restrict__ X,
    const float* __restrict__ Wf, const float* __restrict__ bf, const float* __restrict__ thf,
    const float* __restrict__ Wmf, const float* __restrict__ bmf,
    const float* __restrict__ Wi, const float* __restrict__ bi, const float* __restrict__ thi,
    const float* __restrict__ Wmi, const float* __restrict__ bmi,
    const float* __restrict__ Wu, const float* __restrict__ bu, const float* __restrict__ thu,
    const float* __restrict__ Wmu, const float* __restrict__ bmu,
    const float* __restrict__ Wo, const float* __restrict__ bo, const float* __restrict__ tho,
    const float* __restrict__ Wmo, const float* __restrict__ bmo,
    float* __restrict__ out)
{
  __shared__ __align__(16) float sh_h[16 * HPAD];     // h state tile  (33 KB)
  __shared__ __align__(16) float sh_part[8 * 256];    // GEMM1 K-chunk partials (8 KB)
  __shared__ __align__(16) float sh_ang[16 * QPAD];   // pre-activation angles
  __shared__ __align__(16) float sh_q[16 * QPAD];     // quantum expectation values

  const int tid  = threadIdx.x;
  const int wv   = tid >> 5;
  const int lane = tid & 31;
  const int nloc = lane & 15;
  const int koff = (lane >> 4) << 1;        // lanes 16-31 hold K=k0+2,k0+3
  const int batchBase = blockIdx.x << 4;

  // GEMM1 wave roles: 2 N-tiles x 4 K-chunks (chunks 0,1: x from global; 2,3: h from LDS)
  const int ntile = wv >> 2;
  const int kc    = wv & 3;

  // Per-lane B-operand pointer for GEMM1 (W row for this lane's output column); time-invariant
  const int ng1 = ntile * 16 + nloc;
  const int g1  = ng1 >> 3;                 // gate 0..3 (f,i,u,o)
  const int r1  = ng1 & 7;                  // qubit row within gate
  const float* Wg = (g1 == 0) ? Wf : (g1 == 1) ? Wi : (g1 == 2) ? Wu : Wo;
  const int kbase = (kc < 2) ? (kc * 256) : (512 + (kc - 2) * 256);
  const float* wptr = Wg + r1 * COMB + kbase + koff;

  // A-operand bases
  const float* xbase = X + (size_t)(batchBase + nloc) * DDIM + (kc & 1) * 256 + koff;
  const int    hoff  = nloc * HPAD + (kc & 1) * 256 + koff;

  // h0 = 0
  for (int i = tid; i < 16 * HPAD; i += TPB) sh_h[i] = 0.0f;

  // c state lives in registers in WMMA C-fragment layout, persistent across time
  float cst[4][8];
  #pragma unroll
  for (int a = 0; a < 4; ++a)
    #pragma unroll
    for (int r = 0; r < 8; ++r) cst[a][r] = 0.0f;

  __syncthreads();

  const size_t OUTH = (size_t)SEQ * BATCH * HDIM;
  const size_t OUTC = OUTH + (size_t)BATCH * HDIM;

  for (int t = 0; t < SEQ; ++t) {
    // ---------------- GEMM1: [x_t, h] @ W_all^T  (K split across 4 waves) ----------------
    v8f acc = {0.f, 0.f, 0.f, 0.f, 0.f, 0.f, 0.f, 0.f};
    if (kc < 2) {
      const float* xp = xbase + (size_t)t * BATCH * DDIM;
      if (t + 1 < SEQ) __builtin_prefetch(xp + (size_t)BATCH * DDIM, 0, 1);
      #pragma unroll 4
      for (int kb = 0; kb < 64; ++kb) {
        v2f a = *(const v2f*)(xp + kb * 4);
        v2f b = *(const v2f*)(wptr + kb * 4);
        acc = __builtin_amdgcn_wmma_f32_16x16x4_f32(false, a, false, b, (short)0, acc, false, false);
      }
    } else {
      #pragma unroll 4
      for (int kb = 0; kb < 64; ++kb) {
        v2f a = *(const v2f*)(&sh_h[hoff + kb * 4]);
        v2f b = *(const v2f*)(wptr + kb * 4);
        acc = __builtin_amdgcn_wmma_f32_16x16x4_f32(false, a, false, b, (short)0, acc, false, false);
      }
    }
    #pragma unroll
    for (int r = 0; r < 8; ++r) sh_part[wv * 256 + r * 32 + lane] = acc[r];
    __syncthreads();

    // ---------------- reduce K-chunk partials, fold in bias + theta ----------------
    #pragma unroll
    for (int it = 0; it < 2; ++it) {
      int idx = tid + it * TPB;          // 512 (M, n_global) pairs
      int M   = idx >> 5;
      int ngi = idx & 31;
      int nn  = ngi >> 4;
      int N   = ngi & 15;
      int rr  = M & 7;
      int ln  = N + ((M & 8) ? 16 : 0);
      float s = sh_part[(nn * 4 + 0) * 256 + rr * 32 + ln]
              + sh_part[(nn * 4 + 1) * 256 + rr * 32 + ln]
              + sh_part[(nn * 4 + 2) * 256 + rr * 32 + ln]
              + sh_part[(nn * 4 + 3) * 256 + rr * 32 + ln];
      int gg = ngi >> 3, wq = ngi & 7;
      const float* bb = (gg == 0) ? bf  : (gg == 1) ? bi  : (gg == 2) ? bu  : bo;
      const float* tt = (gg == 0) ? thf : (gg == 1) ? thi : (gg == 2) ? thu : tho;
      sh_ang[M * QPAD + ngi] = s + bb[wq] + tt[wq];
    }
    __syncthreads();

    // ---------------- quantum gate: cos products over CNOT-ring Z-string masks ----------------
    // masks: w=0 -> bits {1..7}; w>=1 -> prefix bits {0..w}
    if (tid < 64) {                      // waves 0-1 fully active
      int Mr = tid >> 2, gg = tid & 3;
      float cv[8];
      #pragma unroll
      for (int i = 0; i < 8; ++i) cv[i] = cosf(sh_ang[Mr * QPAD + gg * 8 + i]);
      float suf = cv[1];
      #pragma unroll
      for (int i = 2; i < 8; ++i) suf *= cv[i];
      sh_q[Mr * QPAD + gg * 8 + 0] = suf;
      float pref = cv[0];
      #pragma unroll
      for (int w = 1; w < 8; ++w) { pref *= cv[w]; sh_q[Mr * QPAD + gg * 8 + w] = pref; }
    }
    __syncthreads();

    // ---------------- GEMM2 (q @ Wm^T, K=8) + LSTM cell update, all in-register ----------------
    #pragma unroll
    for (int jj = 0; jj < 4; ++jj) {
      const int j   = wv + jj * 8;       // fixed N-tile ownership -> c fragment stays resident
      const int col = j * 16 + nloc;

      auto gate_mm = [&](const float* Wm, const float* bm, int g) -> v8f {
        float bias = bm[col];
        v8f c = {bias, bias, bias, bias, bias, bias, bias, bias};
        #pragma unroll
        for (int k0 = 0; k0 < 8; k0 += 4) {
          v2f a = *(const v2f*)(&sh_q[nloc * QPAD + g * 8 + k0 + koff]);
          v2f b = *(const v2f*)(Wm + col * 8 + k0 + koff);
          c = __builtin_amdgcn_wmma_f32_16x16x4_f32(false, a, false, b, (short)0, c, false, false);
        }
        return c;
      };

      v8f pf = gate_mm(Wmf, bmf, 0);
      v8f pi = gate_mm(Wmi, bmi, 1);
      v8f pu = gate_mm(Wmu, bmu, 2);
      v8f po = gate_mm(Wmo, bmo, 3);

      #pragma unroll
      for (int r = 0; r < 8; ++r) {
        float fg = sigm_(pf[r]);
        float ig = sigm_(pi[r]);
        float ug = tanhf(pu[r]);
        float og = sigm_(po[r]);
        float cc = fg * cst[jj][r] + ig * ug;
        cst[jj][r] = cc;
        float hv = og * tanhf(cc);
        int M = r + ((lane >> 4) << 3);
        out[((size_t)t * BATCH + batchBase + M) * HDIM + col] = hv;
        sh_h[M * HPAD + col] = hv;
      }
    }
    __syncthreads();   // h writes visible before next step's GEMM1 reads
  }

  // ---------------- final h and c ----------------
  for (int i = tid; i < 16 * HDIM; i += TPB) {
    int M = i >> 9, col = i & 511;
    out[OUTH + (size_t)(batchBase + M) * HDIM + col] = sh_h[M * HPAD + col];
  }
  #pragma unroll
  for (int jj = 0; jj < 4; ++jj) {
    const int j   = wv + jj * 8;
    const int col = j * 16 + nloc;
    #pragma unroll
    for (int r = 0; r < 8; ++r) {
      int M = r + ((lane >> 4) << 3);
      out[OUTC + (size_t)(batchBase + M) * HDIM + col] = cst[jj][r];
    }
  }
}

extern "C" void kernel_launch(void* const* d_in, const int* in_sizes, int n_in,
                              void* d_out, int out_size, void* d_ws, size_t ws_size,
                              hipStream_t stream) {
  (void)in_sizes; (void)n_in; (void)out_size; (void)d_ws; (void)ws_size;
  // dict order: inputs, then per gate (f,i,u,o): W, b, theta, Wm, bm
  const float* X = (const float*)d_in[0];
#define G(i) ((const float*)d_in[i])
  qlstm_fused<<<dim3(BATCH / 16), dim3(TPB), 0, stream>>>(
      X,
      G(1),  G(2),  G(3),  G(4),  G(5),    // forget
      G(6),  G(7),  G(8),  G(9),  G(10),   // input
      G(11), G(12), G(13), G(14), G(15),   // update
      G(16), G(17), G(18), G(19), G(20),   // output
      (float*)d_out);
#undef G
}